// ChromDecoder_17935783428336
// MI455X (gfx1250) — compile-verified
//
#include <hip/hip_runtime.h>
#include <hip/hip_bf16.h>
#include <math.h>

#define B_   2048
#define LAT  16
#define C_   23
#define HID0 16
#define HID1 32
#define NOUT 4000
#define J1   (C_*HID0)   /* 368 */
#define J2   (C_*HID1)   /* 736 */
#define EPSV 1e-5f

typedef float v2f __attribute__((ext_vector_type(2)));
typedef float v8f __attribute__((ext_vector_type(8)));

__device__ __forceinline__ float leaky(float v) { return fmaxf(v, 0.2f * v); }

// ---------------- Stage 1: h = x @ W1^T + b1  (2048 x 368, K=16) ----------------
__global__ void k1_gemm1(const float* __restrict__ x, const float* __restrict__ W1,
                         const float* __restrict__ b1, float* __restrict__ h) {
  int t = blockIdx.x * 256 + threadIdx.x;
  if (t >= B_ * J1) return;
  int b = t / J1, j = t - b * J1;
  const float* xr = x + (size_t)b * LAT;
  const float* wr = W1 + (size_t)j * LAT;
  float acc = b1[j];
#pragma unroll
  for (int i = 0; i < LAT; ++i) acc = fmaf(xr[i], wr[i], acc);
  h[t] = acc;
}

// ------------- BatchNorm stats per column -> folded scale/shift ------------------
// grid = J blocks, 256 threads; biased variance (matches torch BN forward)
__global__ void k2_bnstats(const float* __restrict__ src, int J,
                           const float* __restrict__ g, const float* __restrict__ beta,
                           float* __restrict__ scale, float* __restrict__ shift) {
  __shared__ float s1[256];
  __shared__ float s2[256];
  int j = blockIdx.x;
  float a = 0.f, b = 0.f;
  for (int r = threadIdx.x; r < B_; r += 256) {
    float v = src[(size_t)r * J + j];
    a += v; b += v * v;
  }
  s1[threadIdx.x] = a; s2[threadIdx.x] = b;
  __syncthreads();
  for (int off = 128; off > 0; off >>= 1) {
    if (threadIdx.x < off) {
      s1[threadIdx.x] += s1[threadIdx.x + off];
      s2[threadIdx.x] += s2[threadIdx.x + off];
    }
    __syncthreads();
  }
  if (threadIdx.x == 0) {
    float mu  = s1[0] * (1.f / B_);
    float var = s2[0] * (1.f / B_) - mu * mu;
    float inv = rsqrtf(var + EPSV);
    float sc  = g[j] * inv;
    scale[j] = sc;
    shift[j] = beta[j] - mu * sc;
  }
}

// --- Stage 2: z[b, c*32+o] = sum_i leaky(bn(h[b, c*16+i])) * W0[c,o,i] + b0 ------
__global__ void k3_gemm2(const float* __restrict__ h, const float* __restrict__ sc1,
                         const float* __restrict__ sh1, const float* __restrict__ W0,
                         const float* __restrict__ b0, float* __restrict__ z) {
  int t = blockIdx.x * 256 + threadIdx.x;
  if (t >= B_ * J2) return;
  int b = t / J2, col = t - b * J2;
  int c = col >> 5;
  const float* hr  = h  + (size_t)b * J1 + c * HID0;
  const float* scp = sc1 + c * HID0;
  const float* shp = sh1 + c * HID0;
  const float* wr  = W0 + (size_t)col * HID0;   // W0 viewed as [C*32][16]
  float acc = b0[col];
#pragma unroll
  for (int i = 0; i < HID0; ++i) {
    float hn = leaky(fmaf(hr[i], scp[i], shp[i]));
    acc = fmaf(hn, wr[i], acc);
  }
  z[t] = acc;
}

// ------------------- Apply BN2 + leaky -> zn (A-matrix of stage 3) ---------------
__global__ void k5_apply(const float* __restrict__ z, const float* __restrict__ sc,
                         const float* __restrict__ sh, float* __restrict__ zn) {
  int t = blockIdx.x * 256 + threadIdx.x;
  if (t >= B_ * J2) return;
  int col = t % J2;
  zn[t] = leaky(fmaf(z[t], sc[col], sh[col]));
}

// ------ Stage 3: per-chromosome [2048x32]@[32x4000] + sigmoid via f32 WMMA -------
// Wave tile: 32(M) x 80(N), K=32 -> 8 k-steps of V_WMMA_F32_16X16X4_F32 per pair,
// 10 accumulators (2 M-sub x 5 N-sub) initialized with the output bias.
// Grid: 23 chrom * 64 mg * 50 ng waves. Output stored non-temporally so the
// 754 MB write-once stream doesn't evict L2-resident W2 (11.8 MB) / zn (6 MB).
__global__ __launch_bounds__(256) void k6_wmma(const float* __restrict__ zn,
                                               const float* __restrict__ W2,
                                               const float* __restrict__ b2,
                                               float* __restrict__ out) {
  const int lane = threadIdx.x & 31;
  const int wid  = blockIdx.x * 8 + (threadIdx.x >> 5);
  const int c    = wid / 3200;            // 64*50 wave-tiles per chromosome
  int rem        = wid - c * 3200;
  const int mg   = rem / 50;
  const int ng   = rem - mg * 50;
  const int m0   = mg * 32;
  const int n0   = ng * 80;
  const int half = lane >> 4;             // K-pair selector (ISA f32 A/B layout)
  const int q    = lane & 15;             // M (for A) / N (for B) index

  // A: zn[m0 + s*16 + q][c*32 + kk*4 + 2*half + v]
  const float* a0 = zn + (size_t)(m0 + q) * J2 + c * HID1 + 2 * half;
  const float* a1 = a0 + (size_t)16 * J2;
  // B: W2[c][n0 + t*16 + q][kk*4 + 2*half + v]   (W2 row-major [C,4000,32])
  const float* bp = W2 + ((size_t)c * NOUT + n0 + q) * HID1 + 2 * half;

  // C/D layout: VGPR r at this lane covers (M = r + 8*half, N = q) -> bias
  // depends only on N, i.e. is uniform across all 8 rows: broadcast-init C.
  v8f acc[2][5];
#pragma unroll
  for (int t = 0; t < 5; ++t) {
    const float bias = b2[(size_t)c * NOUT + n0 + t * 16 + q];
#pragma unroll
    for (int s = 0; s < 2; ++s)
#pragma unroll
      for (int r = 0; r < 8; ++r) acc[s][t][r] = bias;
  }

#pragma unroll
  for (int kk = 0; kk < 8; ++kk) {
    const int ko = kk * 4;
    v2f a[2], bf[5];
    a[0] = *(const v2f*)(a0 + ko);
    a[1] = *(const v2f*)(a1 + ko);
#pragma unroll
    for (int t = 0; t < 5; ++t)
      bf[t] = *(const v2f*)(bp + (size_t)t * 16 * HID1 + ko);
#pragma unroll
    for (int s = 0; s < 2; ++s)
#pragma unroll
      for (int t = 0; t < 5; ++t)
        acc[s][t] = __builtin_amdgcn_wmma_f32_16x16x4_f32(
            false, a[s], false, bf[t], (short)0, acc[s][t], false, false);
  }

  // Epilogue: sigmoid (fast rcp) + non-temporal store.
#pragma unroll
  for (int t = 0; t < 5; ++t) {
    const int col = n0 + t * 16 + q;
#pragma unroll
    for (int s = 0; s < 2; ++s) {
#pragma unroll
      for (int r = 0; r < 8; ++r) {
        const int row = m0 + s * 16 + half * 8 + r;
        const float e = __expf(-acc[s][t][r]);
        const float sig = __builtin_amdgcn_rcpf(1.0f + e);
        __builtin_nontemporal_store(
            sig, out + (size_t)row * (C_ * NOUT) + (size_t)c * NOUT + col);
      }
    }
  }
}

extern "C" void kernel_launch(void* const* d_in, const int* in_sizes, int n_in,
                              void* d_out, int out_size, void* d_ws, size_t ws_size,
                              hipStream_t stream) {
  const float* x   = (const float*)d_in[0];
  const float* W1  = (const float*)d_in[1];
  const float* b1  = (const float*)d_in[2];
  const float* g1  = (const float*)d_in[3];
  const float* be1 = (const float*)d_in[4];
  const float* W0  = (const float*)d_in[5];
  const float* b0  = (const float*)d_in[6];
  const float* g0  = (const float*)d_in[7];
  const float* bb0 = (const float*)d_in[8];
  const float* W2  = (const float*)d_in[9];
  const float* b2  = (const float*)d_in[10];
  float* out = (float*)d_out;

  // Workspace layout (floats): h | sc1 | sh1 | z | sc0 | sh0 | zn  (~14.4 MB)
  float* ws  = (float*)d_ws;
  float* h   = ws;
  float* sc1 = h   + (size_t)B_ * J1;
  float* sh1 = sc1 + J1;
  float* z   = sh1 + J1;
  float* sc0 = z   + (size_t)B_ * J2;
  float* sh0 = sc0 + J2;
  float* zn  = sh0 + J2;

  k1_gemm1<<<(B_ * J1 + 255) / 256, 256, 0, stream>>>(x, W1, b1, h);
  k2_bnstats<<<J1, 256, 0, stream>>>(h, J1, g1, be1, sc1, sh1);
  k3_gemm2<<<(B_ * J2 + 255) / 256, 256, 0, stream>>>(h, sc1, sh1, W0, b0, z);
  k2_bnstats<<<J2, 256, 0, stream>>>(z, J2, g0, bb0, sc0, sh0);
  k5_apply<<<(B_ * J2 + 255) / 256, 256, 0, stream>>>(z, sc0, sh0, zn);
  // 23 * 64 * 50 = 73600 waves, 8 waves (256 threads) per block
  k6_wmma<<<(C_ * 64 * 50) / 8, 256, 0, stream>>>(zn, W2, b2, out);
}